// RandomStretchSqueeze_58188216926883
// MI455X (gfx1250) — compile-verified
//
#include <hip/hip_runtime.h>

// RandomStretchSqueeze for MI455X (gfx1250, wave32).
// Memory-bound (≈256 MB traffic -> ~11 us @ 23.3 TB/s). Strategy:
//   - per block: stage the contiguous source span for a 4096-output chunk
//     into LDS via GLOBAL_LOAD_ASYNC_TO_LDS_B128 (ASYNCcnt),
//   - s_wait_asynccnt 0 + workgroup barrier,
//   - bilinear gather from LDS (ds_load_b32, ~unit-stride lanes -> no bank
//     conflicts), non-temporal coalesced stores for the output stream.

#ifndef __has_builtin
#define __has_builtin(x) 0
#endif

#if __has_builtin(__builtin_amdgcn_global_load_async_to_lds_b128)
#define USE_ASYNC_LDS 1
#else
#define USE_ASYNC_LDS 0
#endif

#define THREADS   256
#define CHUNK     4096
// staged floats: CHUNK * max(T/L) (=1.1112) + floor/align margins < 4576
#define SMEM_ELTS 4608

typedef int vi4 __attribute__((vector_size(16)));

__device__ __forceinline__ int imin(int a, int b) { return a < b ? a : b; }
__device__ __forceinline__ int imax(int a, int b) { return a > b ? a : b; }

__global__ __launch_bounds__(THREADS)
void RandomStretchSqueeze_kernel(const float* __restrict__ in,
                                 const float* __restrict__ tss,
                                 float* __restrict__ out, int T)
{
    __shared__ __align__(16) float smem[SMEM_ELTS];

    const int b   = blockIdx.y;
    const int j0  = blockIdx.x * CHUNK;
    const int tid = threadIdx.x;

    // Per-row scalars (match reference exactly):
    // L = trunc(T * f) (>=1); centered crop/pad offset; scale = T/L.
    const float f = tss[b];
    int L = (int)((float)T * f);
    if (L < 1) L = 1;
    const int   off   = (L >= T) ? (L - T) / 2 : -((T - L) / 2);
    const float scale = (float)T / (float)L;

    // Valid resampled-index range hit by this chunk -> conservative src span.
    const int rvlo = imax(j0 + off, 0);
    const int rvhi = imin(j0 + CHUNK - 1 + off, L - 1);
    int s0, span4;
    if (rvhi >= rvlo) {
        s0 = (int)floorf(((float)rvlo + 0.5f) * scale - 0.5f) - 1;
        s0 = imax(s0, 0) & ~3;                       // 16B-align the base
        int s1 = (int)floorf(((float)rvhi + 0.5f) * scale - 0.5f) + 2;
        s1 = imin(s1, T - 1);
        span4 = (s1 - s0 + 4) >> 2;                  // # of 16B chunks
        span4 = imin(imax(span4, 1), SMEM_ELTS / 4);
    } else {                                          // chunk entirely in pad
        s0 = 0;
        span4 = 1;
    }

    const float* gsrc = in + (size_t)b * (size_t)T + (size_t)s0;

#if USE_ASYNC_LDS
    // CDNA5 async copy: 16 bytes per lane directly into LDS, no VGPR staging.
    for (int i = tid; i < span4; i += THREADS) {
        __builtin_amdgcn_global_load_async_to_lds_b128(
            (__attribute__((address_space(1))) vi4*)(gsrc + 4 * i),
            (__attribute__((address_space(3))) vi4*)(&smem[4 * i]),
            0, 0);
    }
#if __has_builtin(__builtin_amdgcn_s_wait_asynccnt)
    __builtin_amdgcn_s_wait_asynccnt(0);
#else
    asm volatile("s_wait_asynccnt 0" ::: "memory");
#endif
#else
    for (int i = tid; i < span4; i += THREADS) {
        *(float4*)&smem[4 * i] = *(const float4*)(gsrc + 4 * i);
    }
#endif
    __syncthreads();

    float* orow = out + (size_t)b * (size_t)T;
    const int spanE = span4 * 4;

#pragma unroll
    for (int k = 0; k < CHUNK / THREADS; ++k) {
        const int j = j0 + tid + k * THREADS;
        const int r = j + off;
        float res = 0.0f;
        if ((unsigned)r < (unsigned)L) {             // r>=0 && r<L
            float x = ((float)r + 0.5f) * scale - 0.5f;
            x = fminf(fmaxf(x, 0.0f), (float)(T - 1));
            const float x0f = floorf(x);
            const float w   = x - x0f;
            const int x0i = (int)x0f;
            const int x1i = imin(x0i + 1, T - 1);
            const int i0 = imin(imax(x0i - s0, 0), spanE - 1);
            const int i1 = imin(imax(x1i - s0, 0), spanE - 1);
            const float g0 = smem[i0];
            const float g1 = smem[i1];
            res = (1.0f - w) * g0 + w * g1;
        }
        __builtin_nontemporal_store(res, &orow[j]);  // write-once stream
    }
}

extern "C" void kernel_launch(void* const* d_in, const int* in_sizes, int n_in,
                              void* d_out, int out_size, void* d_ws, size_t ws_size,
                              hipStream_t stream) {
    const float* in  = (const float*)d_in[0];   // [B, T] fp32
    const float* tss = (const float*)d_in[1];   // [B]    fp32
    float* out = (float*)d_out;                 // [B, T] fp32

    const int n  = in_sizes[0];
    const int Bn = in_sizes[1];
    const int T  = n / Bn;

    dim3 grid((unsigned)((T + CHUNK - 1) / CHUNK), (unsigned)Bn, 1);
    hipLaunchKernelGGL(RandomStretchSqueeze_kernel, grid, dim3(THREADS), 0,
                       stream, in, tss, out, T);
}